// Model_30459908063473
// MI455X (gfx1250) — compile-verified
//
#include <hip/hip_runtime.h>
#include <math.h>

typedef float v2f __attribute__((ext_vector_type(2)));
typedef float v8f __attribute__((ext_vector_type(8)));

#define BATCH 64
#define FEA   18496
#define MEMD  512
#define BN_EPS 1e-5f

#define KSPLIT 17
#define KCHUNK (FEA / KSPLIT)      // 1088, multiple of 4

// ---------------- encoder ----------------

// conv1: 1x1, stride 2, pad 1, 3->16, 128x128 -> 65x65, then BN+ReLU
__global__ void k_enc_conv1(const float* __restrict__ x, const float* __restrict__ w,
                            const float* __restrict__ bias, const float* __restrict__ g,
                            const float* __restrict__ beta, float* __restrict__ out) {
    int idx = blockIdx.x * blockDim.x + threadIdx.x;
    const int total = BATCH * 16 * 65 * 65;
    if (idx >= total) return;
    int ow = idx % 65; int t = idx / 65;
    int oh = t % 65;   t /= 65;
    int oc = t % 16;   int b = t / 16;
    int ih = 2 * oh - 1, iw = 2 * ow - 1;
    float acc = bias[oc];
    if (ih >= 0 && ih < 128 && iw >= 0 && iw < 128) {
        const float* xb = x + (((size_t)b * 3) * 128 + ih) * 128 + iw;
        acc = fmaf(w[oc * 3 + 0], xb[0],             acc);
        acc = fmaf(w[oc * 3 + 1], xb[128 * 128],     acc);
        acc = fmaf(w[oc * 3 + 2], xb[2 * 128 * 128], acc);
    }
    float s = g[oc] * rsqrtf(1.0f + BN_EPS);
    float v = acc * s + beta[oc];
    out[idx] = v > 0.0f ? v : 0.0f;
}

// generic 3x3 stride-2 pad-1 conv + BN + ReLU
__global__ void k_enc_conv3x3(const float* __restrict__ x, const float* __restrict__ w,
                              const float* __restrict__ bias, const float* __restrict__ g,
                              const float* __restrict__ beta, float* __restrict__ out,
                              int Cin, int Hin, int Cout, int Hout) {
    int idx = blockIdx.x * blockDim.x + threadIdx.x;
    int total = BATCH * Cout * Hout * Hout;
    if (idx >= total) return;
    int ow = idx % Hout; int t = idx / Hout;
    int oh = t % Hout;   t /= Hout;
    int oc = t % Cout;   int b = t / Cout;
    float acc = bias[oc];
    for (int ic = 0; ic < Cin; ++ic) {
        const float* xp = x + (((size_t)b * Cin + ic) * Hin) * Hin;
        const float* wp = w + ((size_t)(oc * Cin + ic)) * 9;
        #pragma unroll
        for (int kh = 0; kh < 3; ++kh) {
            int ih = 2 * oh - 1 + kh;
            if (ih < 0 || ih >= Hin) continue;
            #pragma unroll
            for (int kw = 0; kw < 3; ++kw) {
                int iw = 2 * ow - 1 + kw;
                if (iw < 0 || iw >= Hin) continue;
                acc = fmaf(xp[ih * Hin + iw], wp[kh * 3 + kw], acc);
            }
        }
    }
    float s = g[oc] * rsqrtf(1.0f + BN_EPS);
    float v = acc * s + beta[oc];
    out[idx] = v > 0.0f ? v : 0.0f;
}

// ---------------- memory unit ----------------

// row L2 norms: blocks 0..63 -> nz over zf rows, blocks 64..575 -> nw over mem_w rows
__global__ void k_norms(const float* __restrict__ zf, const float* __restrict__ memw,
                        float* __restrict__ nz, float* __restrict__ nw) {
    __shared__ float sm[256];
    int row = blockIdx.x;
    const float* p = (row < BATCH) ? (zf + (size_t)row * FEA)
                                   : (memw + (size_t)(row - BATCH) * FEA);
    float acc = 0.0f;
    for (int i = threadIdx.x; i < FEA; i += 256) { float v = p[i]; acc = fmaf(v, v, acc); }
    sm[threadIdx.x] = acc; __syncthreads();
    for (int s = 128; s > 0; s >>= 1) {
        if (threadIdx.x < s) sm[threadIdx.x] += sm[threadIdx.x + s];
        __syncthreads();
    }
    if (threadIdx.x == 0) {
        float n = sqrtf(sm[0]);
        if (row < BATCH) nz[row] = n; else nw[row - BATCH] = n;
    }
}

// dot partials: zf @ mem_w^T (NT GEMM, M=64 N=512 K=18496), K split 17 ways for
// occupancy (2176 waves instead of 128). 8 waves per 256-thread block; each wave
// owns one 16x16 tile x one K-chunk, accumulating with f32 WMMA 16x16x4.
__global__ void k_dot_partial(const float* __restrict__ zf, const float* __restrict__ memw,
                              float* __restrict__ partial) {
    int wid = blockIdx.x * 8 + (threadIdx.x >> 5);   // 0 .. 2175
    int split = wid / 128;
    int tile  = wid % 128;                           // 4 x 32 tiles
    int mt = tile >> 5;
    int nt = tile & 31;
    int lane = threadIdx.x & 31;
    int lo = lane & 15, hi = lane >> 4;
    int kbase = split * KCHUNK;
    const float* ap = zf   + (size_t)(mt * 16 + lo) * FEA + kbase + 2 * hi;
    const float* bp = memw + (size_t)(nt * 16 + lo) * FEA + kbase + 2 * hi;
    v8f c = {};
    #pragma unroll 4
    for (int k0 = 0; k0 < KCHUNK; k0 += 4) {
        v2f a = *(const v2f*)(ap + k0);
        v2f b = *(const v2f*)(bp + k0);
        c = __builtin_amdgcn_wmma_f32_16x16x4_f32(false, a, false, b, (short)0, c, false, false);
    }
    float* pp = partial + (size_t)split * (BATCH * MEMD);
    int col = nt * 16 + lo;
    #pragma unroll
    for (int v = 0; v < 8; ++v) {
        int row = mt * 16 + v + 8 * hi;
        pp[row * MEMD + col] = c[v];
    }
}

// deterministic K-split reduction + cosine normalization -> logits
__global__ void k_logit_reduce(const float* __restrict__ partial, const float* __restrict__ nz,
                               const float* __restrict__ nw, float* __restrict__ logit) {
    int idx = blockIdx.x * blockDim.x + threadIdx.x;
    if (idx >= BATCH * MEMD) return;
    int row = idx / MEMD, col = idx % MEMD;
    float acc = 0.0f;
    #pragma unroll
    for (int s = 0; s < KSPLIT; ++s) acc += partial[(size_t)s * (BATCH * MEMD) + idx];
    float den = nz[row] * nw[col];
    den = den > 1e-8f ? den : 1e-8f;
    logit[idx] = acc / den;
}

// softmax over 512 + hard-shrink + L1 renorm, one block per batch row
__global__ void k_softmax_shrink(const float* __restrict__ logit, float* __restrict__ mw) {
    __shared__ float sm[256];
    int row = blockIdx.x;
    int tid = threadIdx.x;
    float v0 = logit[row * MEMD + tid];
    float v1 = logit[row * MEMD + tid + 256];
    sm[tid] = fmaxf(v0, v1); __syncthreads();
    for (int s = 128; s > 0; s >>= 1) {
        if (tid < s) sm[tid] = fmaxf(sm[tid], sm[tid + s]);
        __syncthreads();
    }
    float mx = sm[0]; __syncthreads();
    float e0 = expf(v0 - mx), e1 = expf(v1 - mx);
    sm[tid] = e0 + e1; __syncthreads();
    for (int s = 128; s > 0; s >>= 1) {
        if (tid < s) sm[tid] += sm[tid + s];
        __syncthreads();
    }
    float denom = sm[0]; __syncthreads();
    float w0 = e0 / denom, w1 = e1 / denom;
    const float th = 1.0f / (float)MEMD;
    w0 = fmaxf(w0 - th, 0.0f) * w0 / (fabsf(w0 - th) + 1e-12f);
    w1 = fmaxf(w1 - th, 0.0f) * w1 / (fabsf(w1 - th) + 1e-12f);
    sm[tid] = fabsf(w0) + fabsf(w1); __syncthreads();
    for (int s = 128; s > 0; s >>= 1) {
        if (tid < s) sm[tid] += sm[tid + s];
        __syncthreads();
    }
    float l1 = sm[0];
    mw[row * MEMD + tid]       = w0 / l1;
    mw[row * MEMD + tid + 256] = w1 / l1;
}

// rd = mw @ mem_w (NN GEMM, M=64 N=18496 K=512). 4624 tiles, 8 waves/block.
__global__ void k_read(const float* __restrict__ mw, const float* __restrict__ memw,
                       float* __restrict__ rd) {
    int tile = blockIdx.x * 8 + (threadIdx.x >> 5);   // 0 .. 4623
    int nt = tile % 1156;
    int mt = tile / 1156;
    int lane = threadIdx.x & 31;
    int lo = lane & 15, hi = lane >> 4;
    int f = nt * 16 + lo;
    const float* ap = mw + (size_t)(mt * 16 + lo) * MEMD + 2 * hi;
    const float* bp = memw + f;
    v8f c = {};
    #pragma unroll 4
    for (int k0 = 0; k0 < MEMD; k0 += 4) {
        v2f a = *(const v2f*)(ap + k0);
        int kb = k0 + 2 * hi;
        v2f b;
        b.x = bp[(size_t)kb * FEA];
        b.y = bp[(size_t)(kb + 1) * FEA];
        c = __builtin_amdgcn_wmma_f32_16x16x4_f32(false, a, false, b, (short)0, c, false, false);
    }
    #pragma unroll
    for (int v = 0; v < 8; ++v) {
        int row = mt * 16 + v + 8 * hi;
        rd[(size_t)row * FEA + f] = c[v];
    }
}

// ---------------- decoder ----------------

// generic ConvTranspose2d (stride 2, pad 1), weight (Cin, Cout, K, K), optional BN+ReLU
__global__ void k_convT(const float* __restrict__ x, const float* __restrict__ w,
                        const float* __restrict__ bias, const float* __restrict__ g,
                        const float* __restrict__ beta, float* __restrict__ out,
                        int Cin, int Hin, int Cout, int Hout, int Kk, int doBn) {
    int idx = blockIdx.x * blockDim.x + threadIdx.x;
    int total = BATCH * Cout * Hout * Hout;
    if (idx >= total) return;
    int ow = idx % Hout; int t = idx / Hout;
    int oh = t % Hout;   t /= Hout;
    int oc = t % Cout;   int b = t / Cout;
    float acc = bias[oc];
    for (int ic = 0; ic < Cin; ++ic) {
        const float* xp = x + (((size_t)b * Cin + ic) * Hin) * Hin;
        const float* wp = w + ((size_t)(ic * Cout + oc)) * Kk * Kk;
        for (int kh = 0; kh < Kk; ++kh) {
            int th_ = oh + 1 - kh;
            if (th_ < 0 || (th_ & 1)) continue;
            int ih = th_ >> 1;
            if (ih >= Hin) continue;
            for (int kw = 0; kw < Kk; ++kw) {
                int tw = ow + 1 - kw;
                if (tw < 0 || (tw & 1)) continue;
                int iw = tw >> 1;
                if (iw >= Hin) continue;
                acc = fmaf(xp[ih * Hin + iw], wp[kh * Kk + kw], acc);
            }
        }
    }
    if (doBn) {
        float s = g[oc] * rsqrtf(1.0f + BN_EPS);
        acc = acc * s + beta[oc];
        acc = acc > 0.0f ? acc : 0.0f;
    }
    out[idx] = acc;
}

// ---------------- host ----------------

extern "C" void kernel_launch(void* const* d_in, const int* in_sizes, int n_in,
                              void* d_out, int out_size, void* d_ws, size_t ws_size,
                              hipStream_t stream) {
    const float* x      = (const float*)d_in[0];
    const float* ec1_w  = (const float*)d_in[1];
    const float* ec1_b  = (const float*)d_in[2];
    const float* ebn1_g = (const float*)d_in[3];
    const float* ebn1_b = (const float*)d_in[4];
    const float* ec2_w  = (const float*)d_in[5];
    const float* ec2_b  = (const float*)d_in[6];
    const float* ebn2_g = (const float*)d_in[7];
    const float* ebn2_b = (const float*)d_in[8];
    const float* ec3_w  = (const float*)d_in[9];
    const float* ec3_b  = (const float*)d_in[10];
    const float* ebn3_g = (const float*)d_in[11];
    const float* ebn3_b = (const float*)d_in[12];
    const float* mem_w  = (const float*)d_in[13];
    const float* dc1_w  = (const float*)d_in[14];
    const float* dc1_b  = (const float*)d_in[15];
    const float* dbn1_g = (const float*)d_in[16];
    const float* dbn1_b = (const float*)d_in[17];
    const float* dc2_w  = (const float*)d_in[18];
    const float* dc2_b  = (const float*)d_in[19];
    const float* dbn2_g = (const float*)d_in[20];
    const float* dbn2_b = (const float*)d_in[21];
    const float* dc3_w  = (const float*)d_in[22];
    const float* dc3_b  = (const float*)d_in[23];

    float* out = (float*)d_out;
    float* mw  = out + (size_t)BATCH * 3 * 132 * 132;   // mw lives in d_out tail

    // workspace layout (floats), with reuse (peak ~35.7 MB):
    //   [0, 4326400)            z1   -> later rd (1183744) + d1 (2367488)
    //   [4326400, 6556672)      z2   -> later d2 (4596736, extends over z3/norms/logit/partial)
    //   [6556672, 7740416)      z3 (= zf)
    //   [7740416, 7740992)      nz(64) + nw(512)
    //   [7740992, 7773760)      logit (64*512)
    //   [7773760, 8330816)      dot partials (17*64*512)
    float* ws = (float*)d_ws;
    float* z1      = ws;
    float* z2      = ws + 4326400;
    float* z3      = ws + 6556672;
    float* nz      = ws + 7740416;
    float* nw      = nz + 64;
    float* logit   = ws + 7740992;
    float* partial = ws + 7773760;
    float* rd      = ws;               // reuse z1 region
    float* d1      = ws + 1183744;     // after rd, still inside old z1 region
    float* d2      = ws + 4326400;     // reuse z2.. region (z3/logit/partial dead by then)

    const int TB = 256;

    // encoder
    k_enc_conv1<<<(BATCH * 16 * 65 * 65 + TB - 1) / TB, TB, 0, stream>>>(
        x, ec1_w, ec1_b, ebn1_g, ebn1_b, z1);
    k_enc_conv3x3<<<(BATCH * 32 * 33 * 33 + TB - 1) / TB, TB, 0, stream>>>(
        z1, ec2_w, ec2_b, ebn2_g, ebn2_b, z2, 16, 65, 32, 33);
    k_enc_conv3x3<<<(BATCH * 64 * 17 * 17 + TB - 1) / TB, TB, 0, stream>>>(
        z2, ec3_w, ec3_b, ebn3_g, ebn3_b, z3, 32, 33, 64, 17);

    // memory unit
    k_norms<<<BATCH + MEMD, TB, 0, stream>>>(z3, mem_w, nz, nw);
    k_dot_partial<<<(KSPLIT * 128) / 8, TB, 0, stream>>>(z3, mem_w, partial);   // 272 blocks, 2176 waves
    k_logit_reduce<<<(BATCH * MEMD + TB - 1) / TB, TB, 0, stream>>>(partial, nz, nw, logit);
    k_softmax_shrink<<<BATCH, TB, 0, stream>>>(logit, mw);
    k_read<<<(4 * 1156) / 8, TB, 0, stream>>>(mw, mem_w, rd);                   // 578 blocks, 4624 waves

    // decoder
    k_convT<<<(BATCH * 32 * 34 * 34 + TB - 1) / TB, TB, 0, stream>>>(
        rd, dc1_w, dc1_b, dbn1_g, dbn1_b, d1, 64, 17, 32, 34, 3, 1);
    k_convT<<<(BATCH * 16 * 67 * 67 + TB - 1) / TB, TB, 0, stream>>>(
        d1, dc2_w, dc2_b, dbn2_g, dbn2_b, d2, 32, 34, 16, 67, 2, 1);
    k_convT<<<(BATCH * 3 * 132 * 132 + TB - 1) / TB, TB, 0, stream>>>(
        d2, dc3_w, dc3_b, dc3_b, dc3_b, out, 16, 67, 3, 132, 2, 0);
}